// LSTMCell_34626026341075
// MI455X (gfx1250) — compile-verified
//
#include <hip/hip_runtime.h>

typedef __attribute__((ext_vector_type(2))) float v2f;
typedef __attribute__((ext_vector_type(8))) float v8f;
typedef __attribute__((ext_vector_type(4))) int   v4i;

#define TILE_M 128
#define TILE_J 32
#define BK 32
#define LDS_STRIDE 36   // 32 + 4 pad: rows stay 16B-aligned (36*4=144B), banks spread
#define H_DIM 2048
#define I_DIM 2048
#define K_DIM 4096
#define BATCH 4096

#define A_ELEMS (TILE_M * LDS_STRIDE)        // 4608 floats
#define B_ELEMS (4 * TILE_J * LDS_STRIDE)    // 4608 floats
#define BUF_ELEMS (A_ELEMS + B_ELEMS)        // 9216 floats = 36864 B per buffer

#if defined(__AMDGCN__) && __has_builtin(__builtin_amdgcn_global_load_async_to_lds_b128)
#define USE_ASYNC 1
#else
#define USE_ASYNC 0
#endif

#if USE_ASYNC
// 16B global -> LDS async copy (tracked by ASYNCcnt, bypasses VGPRs)
__device__ __forceinline__ void async_copy16(const float* g, float* l) {
    __builtin_amdgcn_global_load_async_to_lds_b128(
        (__attribute__((address_space(1))) v4i*)(g),
        (__attribute__((address_space(3))) v4i*)(l),
        0, 0);
}
#endif

__device__ __forceinline__ void wait_async_done() {
#if USE_ASYNC
#if __has_builtin(__builtin_amdgcn_s_wait_asynccnt)
    __builtin_amdgcn_s_wait_asynccnt(0);
#else
    asm volatile("s_wait_asynccnt 0x0" ::: "memory");
#endif
#endif
}

// Stage one K-slab: A tile (128 x 32) + 4 gate weight tiles (32 x 32 each) into `buf`.
__device__ __forceinline__ void stage_tiles(
    float* __restrict__ buf,
    const float* __restrict__ x, const float* __restrict__ h,
    const float* const* __restrict__ W,
    int mBase, int jBase, int kt, int ld_row, int ld_col4)
{
    // combined = [h | x]; BK divides H_DIM so a tile never straddles the seam
    const float* srcA = (kt < H_DIM) ? (h + (size_t)mBase * H_DIM + kt)
                                     : (x + (size_t)mBase * I_DIM + (kt - H_DIM));
#if USE_ASYNC
#pragma unroll
    for (int t = 0; t < 4; ++t) {
        const int row = ld_row + t * 32;
        async_copy16(srcA + (size_t)row * H_DIM + ld_col4,
                     &buf[row * LDS_STRIDE + ld_col4]);
    }
#pragma unroll
    for (int g = 0; g < 4; ++g) {
        async_copy16(W[g] + (size_t)(jBase + ld_row) * K_DIM + kt + ld_col4,
                     &buf[A_ELEMS + (g * TILE_J + ld_row) * LDS_STRIDE + ld_col4]);
    }
#else
#pragma unroll
    for (int t = 0; t < 4; ++t) {
        const int row = ld_row + t * 32;
        float4 v = *(const float4*)(srcA + (size_t)row * H_DIM + ld_col4);
        *(float4*)&buf[row * LDS_STRIDE + ld_col4] = v;
    }
#pragma unroll
    for (int g = 0; g < 4; ++g) {
        float4 v = *(const float4*)(W[g] + (size_t)(jBase + ld_row) * K_DIM + kt + ld_col4);
        *(float4*)&buf[A_ELEMS + (g * TILE_J + ld_row) * LDS_STRIDE + ld_col4] = v;
    }
#endif
}

__global__ __launch_bounds__(256) void lstm_cell_fused(
    const float* __restrict__ x, const float* __restrict__ h, const float* __restrict__ c,
    const float* __restrict__ Wf, const float* __restrict__ bf,
    const float* __restrict__ Wi, const float* __restrict__ bi,
    const float* __restrict__ Wg, const float* __restrict__ bg,
    const float* __restrict__ Wo, const float* __restrict__ bo,
    float* __restrict__ out)
{
    __shared__ float lds[2 * BUF_ELEMS];   // double-buffered: 73728 B

    const int tid    = threadIdx.x;
    const int lane   = tid & 31;
    const int wv     = tid >> 5;       // 0..7
    const int wave_m = wv & 3;         // 0..3 -> 32-row slab
    const int wave_n = wv >> 2;        // 0..1 -> 16-col half

    const int jBase = blockIdx.x * TILE_J;   // hidden-feature tile
    const int mBase = blockIdx.y * TILE_M;   // batch tile

    const float* W[4] = { Wf, Wi, Wg, Wo };

    v8f zero = {};
    v8f acc[4][2];
#pragma unroll
    for (int g = 0; g < 4; ++g) {
#pragma unroll
        for (int s = 0; s < 2; ++s) acc[g][s] = zero;
    }

    const int ld_row  = tid >> 3;          // 0..31
    const int ld_col4 = (tid & 7) * 4;     // float offset of a float4

    const int mFrag = wave_m * 32 + (lane & 15);
    const int nFrag = wave_n * 16 + (lane & 15);
    const int kLane = (lane >> 4) * 2;     // which K-pair this half-wave owns

    // ---- software pipeline: prefetch slab 0 ----
    stage_tiles(&lds[0], x, h, W, mBase, jBase, 0, ld_row, ld_col4);

    for (int i = 0; i < K_DIM / BK; ++i) {
        float* cur = &lds[(i & 1) * BUF_ELEMS];
        float* nxt = &lds[((i + 1) & 1) * BUF_ELEMS];

        wait_async_done();     // this wave's async copies into `cur` are complete
        __syncthreads();       // all waves' copies visible; all done reading `nxt`'s old data

        const int kt_next = (i + 1) * BK;
        if (kt_next < K_DIM)
            stage_tiles(nxt, x, h, W, mBase, jBase, kt_next, ld_row, ld_col4);

        // ---- 8 k-steps of V_WMMA_F32_16X16X4_F32 on `cur` ----
#pragma unroll
        for (int kk = 0; kk < BK; kk += 4) {
            v2f a0 = *(const v2f*)&cur[(mFrag +  0) * LDS_STRIDE + kk + kLane];
            v2f a1 = *(const v2f*)&cur[(mFrag + 16) * LDS_STRIDE + kk + kLane];
#pragma unroll
            for (int g = 0; g < 4; ++g) {
                v2f b = *(const v2f*)&cur[A_ELEMS + (g * TILE_J + nFrag) * LDS_STRIDE + kk + kLane];
                acc[g][0] = __builtin_amdgcn_wmma_f32_16x16x4_f32(
                    false, a0, false, b, (short)0, acc[g][0], false, false);
                acc[g][1] = __builtin_amdgcn_wmma_f32_16x16x4_f32(
                    false, a1, false, b, (short)0, acc[g][1], false, false);
            }
        }
    }

    // ---- fused epilogue: gates -> (h_t, c_t) straight from accumulators ----
    // C/D layout: lane L, VGPR r -> M = r + 8*(L>>4), N = L&15
    const int jj = jBase + wave_n * 16 + (lane & 15);
    const float bfv = bf[jj], biv = bi[jj], bgv = bg[jj], bov = bo[jj];
    const int mHalf = (lane >> 4) * 8;

#pragma unroll
    for (int s = 0; s < 2; ++s) {
        const int rowBase = mBase + wave_m * 32 + s * 16 + mHalf;
#pragma unroll
        for (int r = 0; r < 8; ++r) {
            const int b_row = rowBase + r;
            const float fz = acc[0][s][r] + bfv;
            const float iz = acc[1][s][r] + biv;
            const float gz = acc[2][s][r] + bgv;
            const float oz = acc[3][s][r] + bov;

            const float ft = 1.0f / (1.0f + __expf(-fz));
            const float it = 1.0f / (1.0f + __expf(-iz));
            const float e2g = __expf(2.0f * gz);
            const float gt = (e2g - 1.0f) / (e2g + 1.0f);
            const float ot = 1.0f / (1.0f + __expf(-oz));

            // single-use streams: keep them out of L2 (reserved for weights/A)
            const float cv = __builtin_nontemporal_load(&c[(size_t)b_row * H_DIM + jj]);
            const float ct = ft * cv + it * gt;
            const float e2c = __expf(2.0f * ct);
            const float th = (e2c - 1.0f) / (e2c + 1.0f);
            const float ht = ot * th;

            __builtin_nontemporal_store(ht, &out[(size_t)b_row * H_DIM + jj]);                          // h_t
            __builtin_nontemporal_store(ct, &out[(size_t)BATCH * H_DIM + (size_t)b_row * H_DIM + jj]);  // c_t
        }
    }
}

extern "C" void kernel_launch(void* const* d_in, const int* in_sizes, int n_in,
                              void* d_out, int out_size, void* d_ws, size_t ws_size,
                              hipStream_t stream) {
    const float* x  = (const float*)d_in[0];
    const float* h  = (const float*)d_in[1];
    const float* c  = (const float*)d_in[2];
    const float* Wf = (const float*)d_in[3];
    const float* bf = (const float*)d_in[4];
    const float* Wi = (const float*)d_in[5];
    const float* bi = (const float*)d_in[6];
    const float* Wg = (const float*)d_in[7];
    const float* bg = (const float*)d_in[8];
    const float* Wo = (const float*)d_in[9];
    const float* bo = (const float*)d_in[10];
    float* out = (float*)d_out;

    dim3 grid(H_DIM / TILE_J, BATCH / TILE_M);   // (64, 32)
    lstm_cell_fused<<<grid, 256, 0, stream>>>(x, h, c, Wf, bf, Wi, bi, Wg, bg, Wo, bo, out);
}